// Synthesizer_6786048328090
// MI455X (gfx1250) — compile-verified
//
#include <hip/hip_runtime.h>
#include <math.h>

#define NSAMP   32768
#define NFRAMES 128
#define NHARM   16
#define FILT    64
#define NNOTES  128
#define BATCH   64
#define NPAIR   (BATCH * NHARM)   // 1024 (b,h) oscillators

typedef __attribute__((ext_vector_type(2))) float v2f;
typedef __attribute__((ext_vector_type(8))) float v8f;

// workspace layout (float offsets)
#define R_OFF   0                                   // 64*16       radians
#define ENV_OFF 1024                                // 64*16*128   env frames
#define FN_OFF  (1024 + BATCH*NHARM*NFRAMES)        // 64*32768    filtered noise * clip(amp)/64

#define PI_F   3.14159265358979323846f
#define TWOPI  6.283185307179586
#define INV2PI 0.15915494309189535

// ---------------- Kernel A: gumbel argmax -> radians, env scan ----------------
__global__ void __launch_bounds__(128) prep_kernel(const float* __restrict__ packed,
                                                   const float* __restrict__ gumbel,
                                                   float* __restrict__ ws) {
    const int b = blockIdx.x;
    const int tid = threadIdx.x;          // 128 threads
    __shared__ float sval[128];
    __shared__ int   sidx[128];
    __shared__ float f0sh;
    const float* pk = packed + b * 480;

    float u = gumbel[b * NNOTES + tid];
    float g = -logf(-logf(u + 1e-10f) + 1e-10f);
    sval[tid] = pk[tid] + g;              // logits + gumbel noise
    sidx[tid] = tid;
    __syncthreads();
    for (int s = 64; s > 0; s >>= 1) {
        if (tid < s) {
            float va = sval[tid], vb = sval[tid + s];
            // argmax, first occurrence wins on tie
            if (vb > va || (vb == va && sidx[tid + s] < sidx[tid])) {
                sval[tid] = vb; sidx[tid] = sidx[tid + s];
            }
        }
        __syncthreads();
    }
    if (tid == 0) {
        int idx = sidx[0];
        float fr = (float)(440.0 * exp2((double)(idx - 69) / 12.0) / 11025.0); // FREQS[idx]
        float f0 = fr * 11025.0f;                          // * NYQ
        f0 = (40.0f / 11025.0f) + f0 * ((3000.0f - 40.0f) / 11025.0f);
        f0sh = f0;
    }
    __syncthreads();
    if (tid < NHARM) {
        const int h = tid;
        ws[R_OFF + b * NHARM + h] = f0sh * (float)(h + 1) / 11025.0f * PI_F;
        // env scan (lax.scan over 128 frames)
        const float ha = pk[384 + h];
        const float hd = 0.1f + pk[400 + h] * 0.9f;
        float cur = 0.f;
        float* env = ws + ENV_OFF + (b * NHARM + h) * NFRAMES;
        for (int f = 0; f < NFRAMES; ++f) {
            float a = pk[256 + f] * 2.f - 1.f;
            cur = fminf(fmaxf(cur + a * ha, 0.f), 1.f);
            env[f] = cur;          // output BEFORE decay
            cur *= hd;
        }
    }
}

// -------- Kernel B: full 32768-pt complex FFT filter chain in 256KB LDS --------
// Exploits CDNA5's 320KB WGP LDS: the whole complex spectrum stays on-chip.
__global__ void __launch_bounds__(1024) fft_kernel(const float* __restrict__ packed,
                                                   const float* __restrict__ noise_u,
                                                   float* __restrict__ ws) {
    extern __shared__ float smem[];
    float* re  = smem;
    float* im  = smem + NSAMP;
    float* fil = smem + 2 * NSAMP;
    const int b = blockIdx.x;
    const int tid = threadIdx.x;
    const float* pk = packed + b * 480;

    if (tid < FILT) fil[tid] = pk[416 + tid];
    for (int i = tid; i < NSAMP; i += 1024) {
        re[i] = noise_u[b * NSAMP + i] * 2.f - 1.f;
        im[i] = 0.f;
    }
    __syncthreads();

    // forward radix-2 DIF: natural in -> bit-reversed out
    for (int lh = 14; lh >= 0; --lh) {
        const int half = 1 << lh;
        for (int idx = tid; idx < NSAMP / 2; idx += 1024) {
            int j  = idx & (half - 1);
            int i0 = ((idx >> lh) << (lh + 1)) | j;
            int i1 = i0 + half;
            float ang = -PI_F * (float)j / (float)half;
            float s, c; __sincosf(ang, &s, &c);
            float ar = re[i0], ai = im[i0], br = re[i1], bi = im[i1];
            re[i0] = ar + br; im[i0] = ai + bi;
            float tr = ar - br, ti = ai - bi;
            re[i1] = tr * c - ti * s;
            im[i1] = tr * s + ti * c;
        }
        __syncthreads();
    }

    // per-bin filter at bit-reversed positions; H Hermitian-symmetric, pad bin = 0
    for (int i = tid; i < NSAMP; i += 1024) {
        unsigned k  = __brev((unsigned)i) >> 17;     // 15-bit reversal
        unsigned ks = min(k, (unsigned)NSAMP - k);
        float H = 0.f;
        if (ks < 16384u) {                           // ks == 16384 -> padded zero
            float c = ((float)ks + 0.5f) * (1.f / 256.f) - 0.5f;
            c = fminf(fmaxf(c, 0.f), 63.f);
            int q0 = (int)floorf(c);
            int q1 = min(q0 + 1, 63);
            float w = c - (float)q0;
            H = fil[q0] * (1.f - w) + fil[q1] * w;
        }
        re[i] *= H; im[i] *= H;
    }
    __syncthreads();

    // inverse radix-2 DIT: bit-reversed in -> natural out (conjugate twiddles)
    for (int lh = 0; lh <= 14; ++lh) {
        const int half = 1 << lh;
        for (int idx = tid; idx < NSAMP / 2; idx += 1024) {
            int j  = idx & (half - 1);
            int i0 = ((idx >> lh) << (lh + 1)) | j;
            int i1 = i0 + half;
            float ang = PI_F * (float)j / (float)half;
            float s, c; __sincosf(ang, &s, &c);
            float ar = re[i0], ai = im[i0], br = re[i1], bi = im[i1];
            float tr = br * c - bi * s;
            float ti = br * s + bi * c;
            re[i0] = ar + tr; im[i0] = ai + ti;
            re[i1] = ar - tr; im[i1] = ai - ti;
        }
        __syncthreads();
    }

    // scale 1/N, apply clip(lerp(amp)), pre-divide by batch for the final mean
    for (int i = tid; i < NSAMP; i += 1024) {
        float val = re[i] * (1.f / (float)NSAMP);
        float c = ((float)i + 0.5f) * (1.f / 256.f) - 0.5f;
        c = fminf(fmaxf(c, 0.f), 127.f);
        int q0 = (int)floorf(c);
        int q1 = min(q0 + 1, 127);
        float w = c - (float)q0;
        float a0 = pk[256 + q0] * 2.f - 1.f;
        float a1 = pk[256 + q1] * 2.f - 1.f;
        float a  = fminf(fmaxf(a0 + w * (a1 - a0), 0.f), 1.f);
        ws[FN_OFF + b * NSAMP + i] = val * a * (1.f / (float)BATCH);
    }
}

// ---- Kernel C: harmonic bank + WMMA 1024-way reduction + noise mix ----
// One wave32 per 16-sample tile (8 tiles / 256-thread block). A = 16x4 ones;
// B columns = time (lane%16), 4 (b,h) rows per chunk; D row0 = per-t sums.
// Env frames + radians staged in LDS: a 128-sample block touches at most
// frames {F, F+1, F+2}, so 3x1024 env values + 1024 radians = 16KB LDS.
__global__ void __launch_bounds__(256) mix_kernel(const float* __restrict__ ws,
                                                  float* __restrict__ out) {
    __shared__ float senv[3 * NPAIR];
    __shared__ float srad[NPAIR];

    const int lane = threadIdx.x & 31;
    const int wave = threadIdx.x >> 5;
    const int tile = blockIdx.x * 8 + wave;
    const int t    = tile * 16 + (lane & 15);

    // base frame for the whole block (uniform)
    const int t0 = blockIdx.x * 128;
    float c0 = ((float)t0 + 0.5f) * (1.f / 256.f) - 0.5f;
    c0 = fminf(fmaxf(c0, 0.f), 127.f);
    const int F = (int)floorf(c0);

    // cooperative stage: env[:, F..F+2] (clamped) and radians
    const float* envws = ws + ENV_OFF;
    for (int p = threadIdx.x; p < NPAIR; p += 256) {
        srad[p] = ws[R_OFF + p];
        #pragma unroll
        for (int j = 0; j < 3; ++j) {
            int f = min(F + j, NFRAMES - 1);
            senv[j * NPAIR + p] = envws[p * NFRAMES + f];
        }
    }
    __syncthreads();

    // env lerp coords for this t (relative to staged frames)
    float cc = ((float)t + 0.5f) * (1.f / 256.f) - 0.5f;
    cc = fminf(fmaxf(cc, 0.f), 127.f);
    const int   q0 = (int)floorf(cc);
    const int   q1 = min(q0 + 1, NFRAMES - 1);
    const float w  = cc - (float)q0;
    const int   j0 = q0 - F;            // 0..1
    const int   j1 = q1 - F;            // 0..2

    const double tphase = (double)(t + 1);

    v8f acc = {0.f, 0.f, 0.f, 0.f, 0.f, 0.f, 0.f, 0.f};
    v2f ones; ones.x = 1.f; ones.y = 1.f;
    const int poff = (lane >> 4) << 1;  // lanes 0-15 -> rows {0,1}; 16-31 -> {2,3}

    for (int p4 = 0; p4 < NPAIR; p4 += 4) {
        int pa = p4 + poff;
        int pb = pa + 1;
        float ra = srad[pa], rb = srad[pb];
        float ea = senv[j0 * NPAIR + pa];
        float eb = senv[j0 * NPAIR + pb];
        ea += w * (senv[j1 * NPAIR + pa] - ea);
        eb += w * (senv[j1 * NPAIR + pb] - eb);
        // branch-free fp64 range reduction: x - trunc(x/2pi)*2pi, then hw sin
        double xa = tphase * (double)ra;
        double xb = tphase * (double)rb;
        float sa = __sinf((float)(xa - trunc(xa * INV2PI) * TWOPI));
        float sb = __sinf((float)(xb - trunc(xb * INV2PI) * TWOPI));
        v2f bm; bm.x = ea * sa; bm.y = eb * sb;
        acc = __builtin_amdgcn_wmma_f32_16x16x4_f32(
            /*neg_a=*/false, ones, /*neg_b=*/false, bm,
            /*c_mod=*/(short)0, acc, /*reuse_a=*/false, /*reuse_b=*/false);
    }

    // noise: sum 64 pre-scaled batches, split across lane halves + shfl_xor(16)
    float part = 0.f;
    const float* fn = ws + FN_OFF + (size_t)(lane >> 4) * 32 * NSAMP;
    for (int bb = 0; bb < 32; ++bb) part += fn[(size_t)bb * NSAMP + t];
    part += __shfl_xor(part, 16, 32);

    if (lane < 16 && t < NSAMP) {
        out[t] = acc[0] * (1.f / (float)NPAIR) + part;
    }
}

extern "C" void kernel_launch(void* const* d_in, const int* in_sizes, int n_in,
                              void* d_out, int out_size, void* d_ws, size_t ws_size,
                              hipStream_t stream) {
    (void)in_sizes; (void)n_in; (void)out_size; (void)ws_size;
    const float* packed = (const float*)d_in[0];
    const float* gumbel = (const float*)d_in[1];
    const float* noise  = (const float*)d_in[2];
    float* out = (float*)d_out;
    float* ws  = (float*)d_ws;

    const int fft_lds = (2 * NSAMP + FILT) * (int)sizeof(float);   // 262,400 B < 320 KB WGP LDS
    hipFuncSetAttribute(reinterpret_cast<const void*>(fft_kernel),
                        hipFuncAttributeMaxDynamicSharedMemorySize, fft_lds);

    prep_kernel<<<BATCH, 128, 0, stream>>>(packed, gumbel, ws);
    fft_kernel<<<BATCH, 1024, fft_lds, stream>>>(packed, noise, ws);
    mix_kernel<<<NSAMP / 128, 256, 0, stream>>>(ws, out);
}